// ScDiVaModel_21895743275761
// MI455X (gfx1250) — compile-verified
//
#include <hip/hip_runtime.h>

typedef __attribute__((ext_vector_type(2))) float v2f;
typedef __attribute__((ext_vector_type(8))) float v8f;

__device__ __forceinline__ float gelu_exact(float x) {
    // exact GELU: 0.5*x*(1+erf(x/sqrt(2)))
    return 0.5f * x * (1.0f + erff(x * 0.70710678118654752440f));
}

#define WMMA_F32(a, b, c) \
    __builtin_amdgcn_wmma_f32_16x16x4_f32(false, (a), false, (b), (short)0, (c), false, false)

// ---------------------------------------------------------------------------
// Wave-level fp32 WMMA GEMM, split-K partial-slab output:
//   part[z][M][N] = A(MxK-chunk z) @ B(K-chunk z x N)
// Block = 128 threads (4 waves). Wave tile 32x64 = 2x4 of 16x16 WMMA tiles,
// 8 x V_WMMA_F32_16X16X4_F32 per K-step of 4. Block tile 32x256.
// Fast path (wave columns fully in-bounds): software-pipelined loop — next
// iteration's A/B fragments are issued before the current WMMAs so the
// s_wait_loadcnt before the WMMA group only waits on the *previous* loads.
// Fragment layouts per CDNA5 ISA 7.12.2 (wave32):
//   A 16x4 f32: lane<16 holds {K0,K1} of row (lane&15); lane>=16 holds {K2,K3}
//   B 4x16 f32: lane<16 holds {K0,K1} of col (lane&15); lane>=16 holds {K2,K3}
//   C 16x16 f32: vgpr j, lanes 0-15 -> row j, lanes 16-31 -> row j+8
// ---------------------------------------------------------------------------
__global__ __launch_bounds__(128)
void gemm_wmma_f32(const float* __restrict__ A, const float* __restrict__ Bm,
                   float* __restrict__ out, int M, int N, int K)
{
    const int lane = threadIdx.x & 31;
    const int wave = threadIdx.x >> 5;
    const int m0 = blockIdx.x * 32;
    const int n0 = blockIdx.y * 256 + wave * 64;

    const int SK   = gridDim.z;
    const int s    = blockIdx.z;
    const int step = (((K + SK - 1) / SK) + 3) & ~3;   // chunk multiple of 4
    const int k0   = s * step;
    const int k1   = (k0 + step < K) ? (k0 + step) : K;
    float* outp = out + (size_t)s * (size_t)M * (size_t)N;

    const int r    = lane & 15;   // row (A) / col (B) within a 16-tile
    const int half = lane >> 4;   // 0 or 1
    const int kb   = half * 2;    // K sub-offset for this lane half

    const float* arow0 = A + (size_t)(m0 + r)      * (size_t)K;
    const float* arow1 = A + (size_t)(m0 + 16 + r) * (size_t)K;

    int cn[4];
#pragma unroll
    for (int t = 0; t < 4; ++t) {
        int c = n0 + t * 16 + r;
        cn[t] = (c < N) ? c : (N - 1);   // clamp loads; stores are guarded
    }

    v8f acc[8];   // [i*4+t], i = M-tile, t = N-tile
    const v8f vzero = {0.f,0.f,0.f,0.f,0.f,0.f,0.f,0.f};
#pragma unroll
    for (int i = 0; i < 8; ++i) acc[i] = vzero;

    // one guarded K-step (used for the K%4 tail and the out-of-bounds-N path)
    auto kstep_guarded = [&](int k) {
        const bool q0 = (k + kb)     < k1;
        const bool q1 = (k + kb + 1) < k1;
        v2f a0, a1, bf[4];
        a0.x = q0 ? arow0[k + kb]     : 0.f;
        a0.y = q1 ? arow0[k + kb + 1] : 0.f;
        a1.x = q0 ? arow1[k + kb]     : 0.f;
        a1.y = q1 ? arow1[k + kb + 1] : 0.f;
#pragma unroll
        for (int t = 0; t < 4; ++t) {
            bf[t].x = q0 ? Bm[(size_t)(k + kb)     * N + cn[t]] : 0.f;
            bf[t].y = q1 ? Bm[(size_t)(k + kb + 1) * N + cn[t]] : 0.f;
        }
#pragma unroll
        for (int t = 0; t < 4; ++t) {
            acc[t]     = WMMA_F32(a0, bf[t], acc[t]);
            acc[4 + t] = WMMA_F32(a1, bf[t], acc[4 + t]);
        }
    };

    const int full = (k1 - k0) & ~3;
    const int kf   = k0 + full;

    if (n0 + 64 <= N) {
        // -------- fast path: immediate-offset B loads + prefetch pipeline ---
        const float* ap0 = arow0 + k0 + kb;
        const float* ap1 = arow1 + k0 + kb;
        const float* bp0 = Bm + (size_t)(k0 + kb) * (size_t)N + (n0 + r);
        const float* bp1 = bp0 + N;
        const size_t bstep = (size_t)4 * (size_t)N;
        const int iters = full >> 2;
        if (iters > 0) {
            v2f ca0 = *(const v2f*)ap0;           // contiguous b64 load
            v2f ca1 = *(const v2f*)ap1;
            v2f cb[4];
#pragma unroll
            for (int t = 0; t < 4; ++t) { cb[t].x = bp0[t * 16]; cb[t].y = bp1[t * 16]; }
            for (int i = 1; i < iters; ++i) {
                ap0 += 4; ap1 += 4; bp0 += bstep; bp1 += bstep;
                v2f na0 = *(const v2f*)ap0;       // prefetch next fragments
                v2f na1 = *(const v2f*)ap1;
                v2f nb[4];
#pragma unroll
                for (int t = 0; t < 4; ++t) { nb[t].x = bp0[t * 16]; nb[t].y = bp1[t * 16]; }
#pragma unroll
                for (int t = 0; t < 4; ++t) {     // consume current fragments
                    acc[t]     = WMMA_F32(ca0, cb[t], acc[t]);
                    acc[4 + t] = WMMA_F32(ca1, cb[t], acc[4 + t]);
                }
                ca0 = na0; ca1 = na1;
#pragma unroll
                for (int t = 0; t < 4; ++t) cb[t] = nb[t];
            }
#pragma unroll
            for (int t = 0; t < 4; ++t) {         // drain last fragments
                acc[t]     = WMMA_F32(ca0, cb[t], acc[t]);
                acc[4 + t] = WMMA_F32(ca1, cb[t], acc[4 + t]);
            }
        }
        if (kf < k1) kstep_guarded(kf);           // K tail (K=41818 -> 2)
    } else {
        // -------- general path (column-clamped); head GEMM edges only ------
        for (int k = k0; k < k1; k += 4) kstep_guarded(k);
    }

    const int rowoff = half * 8;
#pragma unroll
    for (int i = 0; i < 2; ++i)
#pragma unroll
    for (int t = 0; t < 4; ++t) {
        const int gn = n0 + t * 16 + r;
        if (gn >= N) continue;
#pragma unroll
        for (int j = 0; j < 8; ++j) {
            const int gm = m0 + i * 16 + rowoff + j;
            if (gm >= M) continue;
            outp[(size_t)gm * N + gn] = acc[i * 4 + t][j];
        }
    }
}

// elementwise epilogue: out = [gelu](sum_z part[z] + bias)
__global__ void epi_elem_k(const float* __restrict__ part, int SK,
                           const float* __restrict__ bias, int doGelu,
                           float* __restrict__ out, int M, int N)
{
    size_t idx = (size_t)blockIdx.x * blockDim.x + threadIdx.x;
    size_t total = (size_t)M * N;
    if (idx >= total) return;
    float a = 0.f;
    for (int s = 0; s < SK; ++s) a += part[(size_t)s * total + idx];
    a += bias[idx % (size_t)N];
    out[idx] = doGelu ? gelu_exact(a) : a;
}

// fused epilogue: out = LayerNorm(sum_z part[z] + bias [+ res]) * g + b
// wave-per-row (wave32), biased variance to match jnp.var. Requires N <= 512.
__global__ __launch_bounds__(128)
void epi_res_ln_k(const float* __restrict__ part, int SK,
                  const float* __restrict__ bias, const float* __restrict__ res,
                  const float* __restrict__ g, const float* __restrict__ bn,
                  float* __restrict__ out, int M, int N)
{
    const int wave = threadIdx.x >> 5, lane = threadIdx.x & 31;
    const int row = blockIdx.x * 4 + wave;
    if (row >= M) return;
    const size_t MN = (size_t)M * N;
    const size_t base = (size_t)row * N;
    float vals[16];
    float s = 0.f, ss = 0.f;
#pragma unroll
    for (int i = 0; i < 16; ++i) {
        const int c = lane + i * 32;
        float v = 0.f;
        if (c < N) {
            for (int sk = 0; sk < SK; ++sk) v += part[(size_t)sk * MN + base + c];
            v += bias[c];
            if (res) v += res[base + c];
        }
        vals[i] = v;
        s += v; ss += v * v;
    }
#pragma unroll
    for (int off = 16; off; off >>= 1) {
        s  += __shfl_xor(s,  off, 32);
        ss += __shfl_xor(ss, off, 32);
    }
    const float mean = s / (float)N;
    const float var  = ss / (float)N - mean * mean;
    const float rs   = rsqrtf(var + 1e-5f);
#pragma unroll
    for (int i = 0; i < 16; ++i) {
        const int c = lane + i * 32;
        if (c < N) out[base + c] = (vals[i] - mean) * rs * g[c] + bn[c];
    }
}

// z = mu + eps * exp(0.5*logvar)
__global__ void reparam_k(const float* __restrict__ mu, const float* __restrict__ lv,
                          const float* __restrict__ eps, float* __restrict__ z, int n)
{
    int i = blockIdx.x * blockDim.x + threadIdx.x;
    if (i < n) z[i] = mu[i] + eps[i] * expf(0.5f * lv[i]);
}

extern "C" void kernel_launch(void* const* d_in, const int* in_sizes, int n_in,
                              void* d_out, int out_size, void* d_ws, size_t ws_size,
                              hipStream_t stream)
{
    (void)in_sizes; (void)n_in; (void)out_size;
    constexpr int Bq = 1024, HID = 512, FF = 2048, L = 12, G = 41818,
                  LAT = 128, NCT = 100;

    // setup_inputs() insertion order, nested dicts flattened in place:
    const float* gene   = (const float*)d_in[0];
    const float* eps    = (const float*)d_in[1];
    const float* W_gene = (const float*)d_in[2];
    const float* b_gene = (const float*)d_in[3];
    const float* ln_e_g = (const float*)d_in[4];
    const float* ln_e_b = (const float*)d_in[5];
    // layers: Wq[6] bq[7] Wk[8] bk[9] unused (S==1 -> softmax over 1 elem == 1)
    const float* Wv   = (const float*)d_in[10];
    const float* bv   = (const float*)d_in[11];
    const float* Wo   = (const float*)d_in[12];
    const float* bo   = (const float*)d_in[13];
    const float* ln1g = (const float*)d_in[14];
    const float* ln1b = (const float*)d_in[15];
    const float* W1   = (const float*)d_in[16];
    const float* b1   = (const float*)d_in[17];
    const float* W2   = (const float*)d_in[18];
    const float* b2   = (const float*)d_in[19];
    const float* ln2g = (const float*)d_in[20];
    const float* ln2b = (const float*)d_in[21];
    const float* W_mu = (const float*)d_in[22];
    const float* b_mu = (const float*)d_in[23];
    const float* W_lv = (const float*)d_in[24];
    const float* b_lv = (const float*)d_in[25];
    const float* W_a1 = (const float*)d_in[26];
    const float* b_a1 = (const float*)d_in[27];
    const float* W_a2 = (const float*)d_in[28];
    const float* b_a2 = (const float*)d_in[29];

    const size_t BH = (size_t)Bq * HID;   // 524288 floats
    const size_t BF = (size_t)Bq * FF;    // 2097152 floats
    const size_t BL = (size_t)Bq * LAT;

    // split-K tiers sized to available workspace (deterministic per ws_size)
    const size_t otherF = 3 * BH /*h,v,attn*/ + BF /*ff1*/ + 3 * BL + BH /*hid*/;
    const size_t availF = ws_size / sizeof(float);
    int SKE, SKW1, SKB;            // embed, FF-expand, generic (K=512/2048)
    size_t Pfloats;
    if (availF >= otherF + 16 * BH)      { SKE = 16; SKW1 = 4; SKB = 8; Pfloats = 16 * BH; }
    else if (availF >= otherF + 8 * BH)  { SKE = 8;  SKW1 = 2; SKB = 8; Pfloats = 8 * BH; }
    else                                 { SKE = 4;  SKW1 = 1; SKB = 4; Pfloats = 4 * BH; }
    const int SKHID = 4;                 // hid GEMM: K=128 -> chunks of 32

    float* ws = (float*)d_ws;
    size_t off = 0;
    float* P    = ws + off; off += Pfloats;   // split-K partial slabs (reused)
    float* h    = ws + off; off += BH;
    float* vbuf = ws + off; off += BH;
    float* attn = ws + off; off += BH;
    float* ff1  = ws + off; off += BF;
    float* mu   = ws + off; off += BL;
    float* lv   = ws + off; off += BL;
    float* zb   = ws + off; off += BL;
    float* hid  = ws + off; off += BH;

    const dim3 blk(128);
    auto gemm = [&](const float* A, const float* Bm, float* out,
                    int M, int N, int K, int SK) {
        dim3 grid((M + 31) / 32, (N + 255) / 256, SK);
        gemm_wmma_f32<<<grid, blk, 0, stream>>>(A, Bm, out, M, N, K);
    };
    auto epiE = [&](int SK, const float* bias, int doGelu, float* out, int M, int N) {
        int total = M * N;
        epi_elem_k<<<(total + 255) / 256, 256, 0, stream>>>(P, SK, bias, doGelu,
                                                            out, M, N);
    };
    auto epiLN = [&](int SK, const float* bias, const float* res, const float* g,
                     const float* bn, float* out, int M, int N) {
        epi_res_ln_k<<<(M + 3) / 4, 128, 0, stream>>>(P, SK, bias, res, g, bn,
                                                      out, M, N);
    };

    // --- embedding: split-K GEMM + fused (reduce + bias + LN) ---
    gemm(gene, W_gene, P, Bq, HID, G, SKE);
    epiLN(SKE, b_gene, nullptr, ln_e_g, ln_e_b, h, Bq, HID);

    // --- transformer layers (S==1: attention == value projection) ---
    for (int l = 0; l < L; ++l) {
        const size_t wsq = (size_t)HID * HID, wff = (size_t)HID * FF;
        const float* Wv_l = Wv + l * wsq;  const float* bv_l = bv + l * HID;
        const float* Wo_l = Wo + l * wsq;  const float* bo_l = bo + l * HID;
        const float* W1_l = W1 + l * wff;  const float* b1_l = b1 + l * FF;
        const float* W2_l = W2 + l * wff;  const float* b2_l = b2 + l * HID;

        gemm(h, Wv_l, P, Bq, HID, HID, SKB);
        epiE(SKB, bv_l, 0, vbuf, Bq, HID);
        gemm(vbuf, Wo_l, P, Bq, HID, HID, SKB);
        epiLN(SKB, bo_l, h, ln1g + l * HID, ln1b + l * HID, attn, Bq, HID);
        gemm(attn, W1_l, P, Bq, FF, HID, SKW1);
        epiE(SKW1, b1_l, 1, ff1, Bq, FF);                       // GELU
        gemm(ff1, W2_l, P, Bq, HID, FF, SKB);
        epiLN(SKB, b2_l, attn, ln2g + l * HID, ln2b + l * HID, h, Bq, HID);
    }

    // --- VAE head ---
    gemm(h, W_mu, P, Bq, LAT, HID, SKB);
    epiE(SKB, b_mu, 0, mu, Bq, LAT);
    gemm(h, W_lv, P, Bq, LAT, HID, SKB);
    epiE(SKB, b_lv, 0, lv, Bq, LAT);
    reparam_k<<<(Bq * LAT + 255) / 256, 256, 0, stream>>>(mu, lv, eps, zb, Bq * LAT);
    gemm(zb, W_a1, P, Bq, HID, LAT, SKHID);
    epiE(SKHID, b_a1, 1, hid, Bq, HID);                          // GELU
    gemm(hid, W_a2, P, Bq, NCT, HID, SKB);
    epiE(SKB, b_a2, 0, (float*)d_out, Bq, NCT);
}